// GATWithEdgeFeatures_8409545966423
// MI455X (gfx1250) — compile-verified
//
#include <hip/hip_runtime.h>

#define IN_CH 256
#define HC 128
#define NH 4
#define CC 32
#define ED 64
#define XS 260   // padded LDS row stride (floats) for x tile: 260 mod 64 banks = 4 -> conflict-free A reads

typedef __attribute__((ext_vector_type(2))) float v2f;
typedef __attribute__((ext_vector_type(8))) float v8f;

// monotonic float -> uint mapping so atomicMax(u32) implements float max (handles negatives)
__device__ __forceinline__ unsigned mapOrd(float f) {
  unsigned u = __float_as_uint(f);
  return (u & 0x80000000u) ? ~u : (u | 0x80000000u);
}
__device__ __forceinline__ float unmapOrd(unsigned m) {
  return (m & 0x80000000u) ? __uint_as_float(m ^ 0x80000000u) : __uint_as_float(~m);
}

// ---------------- dense GEMM: x_proj = x @ W  (fp32 WMMA 16x16x4) ----------------
// Block = 256 threads (8 waves). Tile = 128 rows x 128 cols. Whole W staged in LDS,
// pre-swizzled so each B fragment (K,K+1 pair per column) is one aligned ds_load_b64:
//   ldsW[(k>>1)*HC*2 + col*2 + (k&1)] = W[k*HC + col]
__global__ void gemm_xproj(const float* __restrict__ x, const float* __restrict__ W,
                           float* __restrict__ xproj, int n) {
  extern __shared__ float lds[];
  float* ldsX = lds;                    // 128 x XS (padded)
  float* ldsW = lds + 128 * XS;         // swizzled 128 pairs x 128 cols x 2
  const int tid  = threadIdx.x;
  const int wave = tid >> 5;
  const int lane = tid & 31;
  const int rowBase = blockIdx.x * 128;

  // stage W pre-swizzled: read row-pairs as float2, interleave, store float4 (b128)
  const float2* W2 = (const float2*)W;  // W2[k*64 + cp] = W[k][2cp..2cp+1]
#pragma unroll
  for (int i = 0; i < 32; ++i) {
    int fi = i * 256 + tid;             // 8192 float4 stores
    int p  = fi >> 6;                   // K-pair index (k = 2p, 2p+1)
    int cp = fi & 63;                   // column-pair index (cols 2cp, 2cp+1)
    float2 r0 = W2[(2 * p) * 64 + cp];
    float2 r1 = W2[(2 * p + 1) * 64 + cp];
    float4 v = {r0.x, r1.x, r0.y, r1.y};
    *(float4*)(ldsW + (size_t)fi * 4) = v;
  }
  // stage x tile: 128x256 floats = 8192 float4, padded rows in LDS
  const float4* x4 = (const float4*)x;
#pragma unroll
  for (int i = 0; i < 32; ++i) {
    int fi = i * 256 + tid;
    int r  = fi >> 6;                   // 64 float4 per source row
    int cq = fi & 63;
    int gr = rowBase + r;
    float4 v = {0.f, 0.f, 0.f, 0.f};
    if (gr < n) v = x4[(size_t)gr * 64 + cq];
    *(float4*)(ldsX + r * XS + cq * 4) = v;
  }
  __syncthreads();

  const int half  = lane >> 4;          // 0: K+0/K+1   1: K+2/K+3
  const int l16   = lane & 15;
  const int rowIn = wave * 16 + l16;    // A-fragment row for this lane

  v8f acc[8] = {};
  for (int k = 0; k < IN_CH; k += 4) {
    int ka = k + half * 2;                                    // even
    v2f a = *(const v2f*)(ldsX + rowIn * XS + ka);            // one ds_load_b64
    const float* bbase = ldsW + (ka >> 1) * (HC * 2) + l16 * 2;
#pragma unroll
    for (int nt = 0; nt < 8; ++nt) {
      v2f b = *(const v2f*)(bbase + nt * 32);                 // one ds_load_b64, imm offset
      acc[nt] = __builtin_amdgcn_wmma_f32_16x16x4_f32(
          false, a, false, b, (short)0, acc[nt], false, false);
    }
  }

  // D layout: VGPR r -> lanes0-15 M=r, lanes16-31 M=8+r
#pragma unroll
  for (int nt = 0; nt < 8; ++nt) {
#pragma unroll
    for (int r = 0; r < 8; ++r) {
      int grow = rowBase + wave * 16 + r + half * 8;
      if (grow < n) xproj[(size_t)grow * HC + nt * 16 + l16] = acc[nt][r];
    }
  }
}

// ---------------- fold u[d,h] = sum_c W_edge[d, h*C+c] * att_edge[h,c] ----------------
__global__ void fold_u(const float* __restrict__ We, const float* __restrict__ attE,
                       float* __restrict__ u) {
  int t = threadIdx.x;          // 256 = ED * NH
  int d = t >> 2, h = t & 3;
  float s = 0.f;
#pragma unroll
  for (int c = 0; c < CC; ++c) s += We[d * HC + h * CC + c] * attE[h * CC + c];
  u[t] = s;                     // u[d*4 + h]
}

// ---------------- per-node attention scalars a_src / a_dst ----------------
__global__ void node_att(const float* __restrict__ xproj, const float* __restrict__ attS,
                         const float* __restrict__ attD, float* __restrict__ aS,
                         float* __restrict__ aD, int n) {
  int t = blockIdx.x * blockDim.x + threadIdx.x;
  if (t >= n * NH) return;
  int h = t & 3;
  int node = t >> 2;
  const float* xp = xproj + (size_t)node * HC + h * CC;
  const float* as = attS + h * CC;
  const float* ad = attD + h * CC;
  float s = 0.f, d = 0.f;
#pragma unroll
  for (int c = 0; c < CC; ++c) { float v = xp[c]; s += v * as[c]; d += v * ad[c]; }
  aS[t] = s;
  aD[t] = d;
}

// ---------------- real edges: a_edge, alpha, atomic segment-max, a_edge running sum ----
__global__ void edge_alpha(const int* __restrict__ ei, const float* __restrict__ ea,
                           const float* __restrict__ u, const float* __restrict__ aS,
                           const float* __restrict__ aD, float* __restrict__ alpha,
                           unsigned* __restrict__ m, float* __restrict__ aeSum, int E) {
  __shared__ float uL[ED * NH];
  __shared__ float sumL[NH];
  int tid = threadIdx.x;
  uL[tid] = u[tid];                       // blockDim == 256 == ED*NH
  if (tid < NH) sumL[tid] = 0.f;
  __syncthreads();

  int e = blockIdx.x * blockDim.x + tid;
  if (e < E) {
    float ae[NH] = {0.f, 0.f, 0.f, 0.f};
    const float4* row = (const float4*)(ea + (size_t)e * ED);
#pragma unroll
    for (int q = 0; q < ED / 4; ++q) {
      float4 v = row[q];
      const float* up = uL + q * 16;
#pragma unroll
      for (int h = 0; h < NH; ++h)
        ae[h] += v.x * up[0 * 4 + h] + v.y * up[1 * 4 + h] +
                 v.z * up[2 * 4 + h] + v.w * up[3 * 4 + h];
    }
    int s = ei[e], d = ei[E + e];
#pragma unroll
    for (int h = 0; h < NH; ++h) {
      float al = aS[s * 4 + h] + aD[d * 4 + h] + ae[h];
      al = (al > 0.f) ? al : 0.2f * al;            // leaky_relu(0.2)
      alpha[(size_t)e * 4 + h] = al;
      atomicMax(&m[d * 4 + h], mapOrd(al));
      atomicAdd(&sumL[h], ae[h]);                  // LDS float atomic (ds_add_f32)
    }
  }
  __syncthreads();
  if (tid < NH) atomicAdd(&aeSum[tid], sumL[tid]);
}

__global__ void fold_loop(const float* __restrict__ aeSum, float* __restrict__ aeLoop, int E) {
  if (threadIdx.x < NH) aeLoop[threadIdx.x] = aeSum[threadIdx.x] / (float)E;
}

// ---------------- self loops: alpha + segment-max ----------------
__global__ void self_alpha(const float* __restrict__ aS, const float* __restrict__ aD,
                           const float* __restrict__ aeLoop, float* __restrict__ alpha,
                           unsigned* __restrict__ m, int n, int E) {
  int node = blockIdx.x * blockDim.x + threadIdx.x;
  if (node >= n) return;
#pragma unroll
  for (int h = 0; h < NH; ++h) {
    float al = aS[node * 4 + h] + aD[node * 4 + h] + aeLoop[h];
    al = (al > 0.f) ? al : 0.2f * al;
    alpha[(size_t)(E + node) * 4 + h] = al;
    atomicMax(&m[node * 4 + h], mapOrd(al));
  }
}

// ---------------- ex = exp(alpha - m[dst]); denom[dst] += ex (in place) ----------------
__global__ void exp_denom(const int* __restrict__ ei, float* __restrict__ alpha,
                          const unsigned* __restrict__ m, float* __restrict__ denom,
                          int E, int n) {
  int t = blockIdx.x * blockDim.x + threadIdx.x;
  if (t >= E + n) return;
  int d = (t < E) ? ei[E + t] : (t - E);
#pragma unroll
  for (int h = 0; h < NH; ++h) {
    float mx = unmapOrd(m[d * 4 + h]);
    float ex = __expf(alpha[(size_t)t * 4 + h] - mx);
    alpha[(size_t)t * 4 + h] = ex;
    atomicAdd(&denom[d * 4 + h], ex);
  }
}

// ---------------- out init & zeroing ----------------
__global__ void init_out(float* __restrict__ out, const float* __restrict__ bias, size_t total) {
  size_t i = (size_t)blockIdx.x * blockDim.x + threadIdx.x;
  if (i < total) out[i] = bias[i & (HC - 1)];
}
__global__ void zero_words(unsigned* __restrict__ p, size_t total) {
  size_t i = (size_t)blockIdx.x * blockDim.x + threadIdx.x;
  if (i < total) p[i] = 0u;
}

// ---------------- weighted scatter: out[dst] += att * x_proj[src] ----------------
// 128 lanes per edge (one per output channel); 2 edges per 256-thread block.
// x_proj (51MB) and out (51MB) fit the 192MB L2 -> gather/scatter stays on-chip.
__global__ void scatter_out(const int* __restrict__ ei, const float* __restrict__ ex,
                            const float* __restrict__ denom, const float* __restrict__ xproj,
                            float* __restrict__ out, int E, int n) {
  int tid = threadIdx.x;
  int j = tid & 127;                         // channel
  long long t = (long long)blockIdx.x * 2 + (tid >> 7);
  if (t >= (long long)E + n) return;
  int s, d;
  if (t < E) { s = ei[t]; d = ei[E + t]; } else { s = (int)(t - E); d = s; }
  int h = j >> 5;
  float w = ex[(size_t)t * 4 + h] / fmaxf(denom[d * 4 + h], 1e-16f);
  atomicAdd(&out[(size_t)d * HC + j], w * xproj[(size_t)s * HC + j]);
}

extern "C" void kernel_launch(void* const* d_in, const int* in_sizes, int n_in,
                              void* d_out, int out_size, void* d_ws, size_t ws_size,
                              hipStream_t stream) {
  const float* x    = (const float*)d_in[0];
  const int*   ei   = (const int*)d_in[1];
  const float* ea   = (const float*)d_in[2];
  const float* W    = (const float*)d_in[3];
  const float* attS = (const float*)d_in[4];
  const float* attD = (const float*)d_in[5];
  const float* We   = (const float*)d_in[6];
  const float* attE = (const float*)d_in[7];
  const float* bias = (const float*)d_in[8];
  float* out = (float*)d_out;

  const int n  = in_sizes[0] / IN_CH;
  const int E  = in_sizes[1] / 2;
  const int Et = E + n;

  // workspace layout (floats); m/denom/aeSum/aeLoop contiguous for one zero pass
  float* ws = (float*)d_ws;
  size_t o = 0;
  float*    xproj  = ws + o;              o += (size_t)n * HC;
  float*    aS     = ws + o;              o += (size_t)n * NH;
  float*    aD     = ws + o;              o += (size_t)n * NH;
  float*    alpha  = ws + o;              o += (size_t)Et * NH;
  unsigned* m      = (unsigned*)(ws + o); o += (size_t)n * NH;
  float*    denom  = ws + o;              o += (size_t)n * NH;
  float*    aeSum  = ws + o;              o += NH;
  float*    aeLoop = ws + o;              o += NH;
  float*    u      = ws + o;              o += ED * NH;

  size_t zwords = (size_t)n * NH * 2 + 2 * NH;   // m + denom + aeSum + aeLoop
  zero_words<<<(unsigned)((zwords + 255) / 256), 256, 0, stream>>>(m, zwords);

  size_t outWords = (size_t)n * HC;
  init_out<<<(unsigned)((outWords + 255) / 256), 256, 0, stream>>>(out, bias, outWords);

  fold_u<<<1, 256, 0, stream>>>(We, attE, u);

  const size_t ldsBytes = (size_t)(128 * XS + IN_CH * HC) * sizeof(float); // ~258 KB of 320 KB/WGP
  gemm_xproj<<<(n + 127) / 128, 256, ldsBytes, stream>>>(x, W, xproj, n);

  node_att<<<((unsigned)n * NH + 255) / 256, 256, 0, stream>>>(xproj, attS, attD, aS, aD, n);
  edge_alpha<<<(E + 255) / 256, 256, 0, stream>>>(ei, ea, u, aS, aD, alpha, m, aeSum, E);
  fold_loop<<<1, 32, 0, stream>>>(aeSum, aeLoop, E);
  self_alpha<<<(n + 255) / 256, 256, 0, stream>>>(aS, aD, aeLoop, alpha, m, n, E);
  exp_denom<<<(Et + 255) / 256, 256, 0, stream>>>(ei, alpha, m, denom, E, n);
  scatter_out<<<(Et + 1) / 2, 256, 0, stream>>>(ei, alpha, denom, xproj, out, E, n);
}